// RelevancePropagationAvgPool2d_22076131901614
// MI455X (gfx1250) — compile-verified
//
#include <hip/hip_runtime.h>
#include <stdint.h>

typedef float v2f __attribute__((ext_vector_type(2)));
typedef float v4f __attribute__((ext_vector_type(4)));

namespace {

constexpr float EPS   = 1e-5f;
constexpr float SIN_T = 1.2246468e-16f;      // float32(np.sin(np.pi)); COS_T == -1 exactly
constexpr float NW    = 1.0f / (2.0f + EPS); // each comp of w/(||w||+eps), ||[1,1,1,1]|| = 2

// 1-ulp hardware rcp/sqrt: error ~1e-7 relative, far inside validation tolerance,
// avoids the ~10-instruction IEEE div/sqrt fixup sequences.
__device__ __forceinline__ float frcp(float x)  { return __builtin_amdgcn_rcpf(x); }
__device__ __forceinline__ float fsqrt(float x) { return __builtin_amdgcn_sqrtf(x); }

// LRP m_rule specialized to w = ones(4), nf = 1 (one 2x2 patch, one relevance scalar)
__device__ __forceinline__ void mrule_patch(float x0, float x1, float x2, float x3,
                                            float rv,
                                            float& y0, float& y1, float& y2, float& y3)
{
    float norm_x = fsqrt(x0*x0 + x1*x1 + x2*x2 + x3*x3) + EPS;
    float inv_nx = frcp(norm_x);
    float n0 = x0 * inv_nx, n1 = x1 * inv_nx, n2 = x2 * inv_nx, n3 = x3 * inv_nx;

    float wx = NW * (n0 + n1 + n2 + n3);
    float o0 = NW - wx * n0;
    float o1 = NW - wx * n1;
    float o2 = NW - wx * n2;
    float o3 = NW - wx * n3;

    float onrm = fsqrt(o0*o0 + o1*o1 + o2*o2 + o3*o3) + EPS;
    float s    = SIN_T * frcp(onrm);  // SIN_T / (||orth||+eps)

    // x_head_k = (COS_T*n_k + SIN_T*north_k)*norm_x = (s*o_k - n_k)*norm_x
    float d0 = x0 - (s * o0 - n0) * norm_x;
    float d1 = x1 - (s * o1 - n1) * norm_x;
    float d2 = x2 - (s * o2 - n2) * norm_x;
    float d3 = x3 - (s * o3 - n3) * norm_x;

    float scale = rv * frcp(d0 + d1 + d2 + d3 + EPS);
    y0 = d0 * scale; y1 = d1 * scale; y2 = d2 * scale; y3 = d3 * scale;
}

// a:[N,224,224], r:[N,112,112], out:[N,224,224], N = 8*64 = 512.
// One thread = TWO adjacent 2x2 patches: (n, oh, 2*ow2) and (n, oh, 2*ow2+1).
// Launch: block (56,4), grid (28, N)  -> exact cover, no divides, no bounds checks.
//   ow2 = threadIdx.x (0..55), oh = 4*blockIdx.x + threadIdx.y (0..111), n = blockIdx.y
__global__ __launch_bounds__(224) void lrp_avgpool2x2_kernel(
    const float* __restrict__ a,
    const float* __restrict__ r,
    float* __restrict__ out)
{
    __shared__ __align__(16) float rbuf[224 * 2];

    const int ow2 = threadIdx.x;
    const int oh  = (blockIdx.x << 2) + threadIdx.y;
    const int n   = blockIdx.y;
    const int tid = threadIdx.y * 56 + threadIdx.x;

    // ---- CDNA5 async global->LDS stage of the two relevance scalars (ASYNCcnt) ----
    const float* gsrc = r + ((n * 112 + oh) * 112 + 2 * ow2);
    // Generic pointer to LDS: low 32 bits are the raw LDS byte offset on amdgcn.
    unsigned ldsoff = (unsigned)(uintptr_t)&rbuf[2 * tid];
    asm volatile("global_load_async_to_lds_b64 %0, %1, off"
                 :: "v"(ldsoff), "v"(gsrc)
                 : "memory");

    // ---- streaming 128-bit non-temporal loads of the two patch rows of `a` ----
    const int base0 = (n * 224 + 2 * oh) * 224 + 4 * ow2;   // 16B aligned
    v4f xr0 = __builtin_nontemporal_load((const v4f*)(a + base0));        // row 2*oh
    v4f xr1 = __builtin_nontemporal_load((const v4f*)(a + base0 + 224));  // row 2*oh+1

    // drain this wave's async copy, then read our own LDS slot (per-lane, no barrier needed)
    asm volatile("s_wait_asynccnt 0" ::: "memory");
    v2f rv = *(const v2f*)&rbuf[2 * tid];

    // patch k-order from unfold: k = ki*2+kj -> (x0,x1) = row0, (x2,x3) = row1
    float a00, a01, a02, a03;   // patch 0 outputs
    float b00, b01, b02, b03;   // patch 1 outputs
    mrule_patch(xr0.x, xr0.y, xr1.x, xr1.y, rv.x, a00, a01, a02, a03);
    mrule_patch(xr0.z, xr0.w, xr1.z, xr1.w, rv.y, b00, b01, b02, b03);

    v4f yr0 = {a00, a01, b00, b01};   // out row 2*oh
    v4f yr1 = {a02, a03, b02, b03};   // out row 2*oh+1

    // fold is a pure placement (stride==kernel, no overlap): NT streaming stores
    __builtin_nontemporal_store(yr0, (v4f*)(out + base0));
    __builtin_nontemporal_store(yr1, (v4f*)(out + base0 + 224));
}

} // anonymous namespace

extern "C" void kernel_launch(void* const* d_in, const int* in_sizes, int n_in,
                              void* d_out, int out_size, void* d_ws, size_t ws_size,
                              hipStream_t stream)
{
    const float* a  = (const float*)d_in[0];  // [8,64,224,224] f32
    const float* r  = (const float*)d_in[1];  // [8,64,112,112] f32
    float* out      = (float*)d_out;          // [8,64,224,224] f32

    int N = in_sizes[0] / (224 * 224);        // 8*64 = 512 images
    dim3 block(56, 4);                        // 224 threads = 7 wave32
    dim3 grid(28, N);                         // 28*4 = 112 patch rows, exact cover
    lrp_avgpool2x2_kernel<<<grid, block, 0, stream>>>(a, r, out);
}